// SparseDepthwiseConv2d_7112465842615
// MI455X (gfx1250) — compile-verified
//
#include <hip/hip_runtime.h>
#include <stdint.h>

// Problem constants (match the reference)
#define BB 4
#define HH 512
#define WW 512
#define CC 256
#define KK 7
#define PP 3
#define TAPS 49         // KK*KK
#define SITES 32        // sites processed per block (multiple of 4)
#define NTHREADS 256    // 8 wave32s

// ---------------------------------------------------------------------------
// Kernel 1: idx_map = -1 everywhere; also transpose weight [C][49] -> wT [49][C]
// so the main kernel's LDS weight table is taps-major (conflict-free b128
// reads, coalesced async copy).
// ---------------------------------------------------------------------------
__global__ __launch_bounds__(NTHREADS) void init_map_and_transpose(
    int* __restrict__ idx_map, const float* __restrict__ weight,
    float* __restrict__ wT, int map_total) {
  int i = blockIdx.x * NTHREADS + threadIdx.x;
  if (i < map_total) idx_map[i] = -1;
  if (i < TAPS * CC) {
    int t = i >> 8;        // i / 256
    int c = i & (CC - 1);  // i % 256
    wT[i] = weight[c * TAPS + t];
  }
}

// ---------------------------------------------------------------------------
// Kernel 2: scatter site ids into the dense (B,H,W) map
// ---------------------------------------------------------------------------
__global__ __launch_bounds__(NTHREADS) void scatter_map(
    const int* __restrict__ idx, int* __restrict__ idx_map, int n) {
  int i = blockIdx.x * NTHREADS + threadIdx.x;
  if (i < n) {
    int b = idx[3 * i + 0];
    int y = idx[3 * i + 1];
    int x = idx[3 * i + 2];
    idx_map[(b * HH + y) * WW + x] = i;
  }
}

// ---------------------------------------------------------------------------
// Kernel 3: main submanifold depthwise conv.
//   Phase 0: async-copy 50 KB transposed weights global->LDS (ASYNCcnt path),
//            overlapped with:
//   Phase A: 32x49 neighbor lookups, compacted into per-site (nb,tap) lists
//            in LDS via LDS atomics (skips the ~80% invalid taps).
//   Phase B: 64 threads per site (4 channels each, float4), 4 sites in flight
//            per block; all traffic is b128, entry loop unrolled x4 so four
//            independent gathers are in flight per wave.
// ---------------------------------------------------------------------------
__global__ __launch_bounds__(NTHREADS) void subm_dwconv7x7(
    const float* __restrict__ feat, const int* __restrict__ idx,
    const float* __restrict__ bias, const int* __restrict__ idx_map,
    const float* __restrict__ wT, float* __restrict__ out, int n) {
  __shared__ float w_lds[TAPS * CC];     // 50176 B, taps-major: w_lds[t*CC + c]
  __shared__ int   list[SITES * TAPS];   // packed (nb<<6 | tap)
  __shared__ int   cnt[SITES];
  __shared__ int   sb[SITES], sy[SITES], sx[SITES];

  const int tid = threadIdx.x;
  const int site0 = blockIdx.x * SITES;

  // ---- Phase 0: issue async bulk copy of weights into LDS (no VGPR staging).
  // 50176 B / 16 B per lane-op = 3136 ops; 256 threads -> ~12 per thread.
  {
    const uint32_t lds_base = (uint32_t)(uintptr_t)(&w_lds[0]); // LDS byte offset
    const uint64_t gbase = (uint64_t)(uintptr_t)wT;
#pragma unroll 1
    for (int k = tid; k < (TAPS * CC * 4) / 16; k += NTHREADS) {
      uint32_t laddr = lds_base + (uint32_t)k * 16u;
      uint64_t gaddr = gbase + (uint64_t)k * 16u;
      asm volatile("global_load_async_to_lds_b128 %0, %1, off"
                   :: "v"(laddr), "v"(gaddr)
                   : "memory");
    }
  }

  // Per-site state init + coordinate preload
  if (tid < SITES) {
    cnt[tid] = 0;
    int i = site0 + tid;
    if (i < n) {
      sb[tid] = idx[3 * i + 0];
      sy[tid] = idx[3 * i + 1];
      sx[tid] = idx[3 * i + 2];
    } else {
      sb[tid] = -1;
    }
  }
  __syncthreads();

  // ---- Phase A: neighbor lookups + compaction (overlaps the async copy)
  for (int k = tid; k < SITES * TAPS; k += NTHREADS) {
    int s = k / TAPS;
    int t = k - s * TAPS;
    int b = sb[s];
    if (b >= 0) {
      int yy = sy[s] + (t / KK) - PP;
      int xx = sx[s] + (t % KK) - PP;
      if (yy >= 0 && yy < HH && xx >= 0 && xx < WW) {
        int nb = idx_map[(b * HH + yy) * WW + xx];
        if (nb >= 0) {
          int pos = atomicAdd(&cnt[s], 1);          // ds_add_rtn_u32
          list[s * TAPS + pos] = (nb << 6) | t;     // nb < 2^18, t < 64
        }
      }
    }
  }

  // Wait for the async weight copy (per-wave ASYNCcnt), then workgroup barrier
#if __has_builtin(__builtin_amdgcn_s_wait_asynccnt)
  __builtin_amdgcn_s_wait_asynccnt(0);
#else
  asm volatile("s_wait_asynccnt 0" ::: "memory");
#endif
  __syncthreads();

  // ---- Phase B: 4 site-groups x 64 threads; 4 channels (float4) per thread
  const int g    = tid >> 6;        // site subgroup 0..3 (2 waves each)
  const int c4   = (tid & 63) << 2; // first of this thread's 4 channels
  const float4 bv = *(const float4*)&bias[c4];

  for (int s = g; s < SITES; s += 4) {
    const int i = site0 + s;
    if (i >= n) continue;
    float4 acc = bv;
    const int m = cnt[s];
    const int* lst = &list[s * TAPS];

    int e = 0;
    // Unrolled x4: 4 independent b128 gathers in flight before the FMAs
    for (; e + 4 <= m; e += 4) {
      const int pk0 = lst[e + 0], pk1 = lst[e + 1];
      const int pk2 = lst[e + 2], pk3 = lst[e + 3];
      const float4 f0 = *(const float4*)&feat[(size_t)(pk0 >> 6) * CC + c4];
      const float4 f1 = *(const float4*)&feat[(size_t)(pk1 >> 6) * CC + c4];
      const float4 f2 = *(const float4*)&feat[(size_t)(pk2 >> 6) * CC + c4];
      const float4 f3 = *(const float4*)&feat[(size_t)(pk3 >> 6) * CC + c4];
      const float4 w0 = *(const float4*)&w_lds[(pk0 & 63) * CC + c4];
      const float4 w1 = *(const float4*)&w_lds[(pk1 & 63) * CC + c4];
      const float4 w2 = *(const float4*)&w_lds[(pk2 & 63) * CC + c4];
      const float4 w3 = *(const float4*)&w_lds[(pk3 & 63) * CC + c4];
      acc.x = fmaf(f0.x, w0.x, acc.x); acc.y = fmaf(f0.y, w0.y, acc.y);
      acc.z = fmaf(f0.z, w0.z, acc.z); acc.w = fmaf(f0.w, w0.w, acc.w);
      acc.x = fmaf(f1.x, w1.x, acc.x); acc.y = fmaf(f1.y, w1.y, acc.y);
      acc.z = fmaf(f1.z, w1.z, acc.z); acc.w = fmaf(f1.w, w1.w, acc.w);
      acc.x = fmaf(f2.x, w2.x, acc.x); acc.y = fmaf(f2.y, w2.y, acc.y);
      acc.z = fmaf(f2.z, w2.z, acc.z); acc.w = fmaf(f2.w, w2.w, acc.w);
      acc.x = fmaf(f3.x, w3.x, acc.x); acc.y = fmaf(f3.y, w3.y, acc.y);
      acc.z = fmaf(f3.z, w3.z, acc.z); acc.w = fmaf(f3.w, w3.w, acc.w);
    }
    for (; e < m; ++e) {
      const int pk = lst[e];
      const float4 f = *(const float4*)&feat[(size_t)(pk >> 6) * CC + c4];
      const float4 w = *(const float4*)&w_lds[(pk & 63) * CC + c4];
      acc.x = fmaf(f.x, w.x, acc.x); acc.y = fmaf(f.y, w.y, acc.y);
      acc.z = fmaf(f.z, w.z, acc.z); acc.w = fmaf(f.w, w.w, acc.w);
    }
    *(float4*)&out[(size_t)i * CC + c4] = acc;  // global_store_b128, coalesced
  }
}

// ---------------------------------------------------------------------------
// Host entry
// ---------------------------------------------------------------------------
extern "C" void kernel_launch(void* const* d_in, const int* in_sizes, int n_in,
                              void* d_out, int out_size, void* d_ws, size_t ws_size,
                              hipStream_t stream) {
  const float* feat   = (const float*)d_in[0];  // N x C f32
  const int*   idx    = (const int*)d_in[1];    // N x 3 i32
  const float* weight = (const float*)d_in[2];  // C x 7 x 7 f32
  const float* bias   = (const float*)d_in[3];  // C f32
  float* out = (float*)d_out;                   // N x C f32
  const int n = in_sizes[1] / 3;

  const int map_total = BB * HH * WW;           // 1,048,576
  int*   idx_map = (int*)d_ws;                                   // 4 MB
  float* wT = (float*)((char*)d_ws + (size_t)map_total * sizeof(int)); // 50 KB

  init_map_and_transpose<<<(map_total + NTHREADS - 1) / NTHREADS, NTHREADS, 0, stream>>>(
      idx_map, weight, wT, map_total);
  scatter_map<<<(n + NTHREADS - 1) / NTHREADS, NTHREADS, 0, stream>>>(idx, idx_map, n);
  subm_dwconv7x7<<<(n + SITES - 1) / SITES, NTHREADS, 0, stream>>>(
      feat, idx, bias, idx_map, wT, out, n);
}